// ProtSiam_52596169506935
// MI455X (gfx1250) — compile-verified
//
#include <hip/hip_runtime.h>
#include <math.h>

#define N_HYPO 384
#define N_HYPER 384
#define HDIM 768
#define NLAB 3

// 1/sqrt(768)
#define INV_SQRT_H 0.03608439182435161f

#if __has_builtin(__builtin_amdgcn_tanhf)
__device__ __forceinline__ float fast_tanh(float x) { return __builtin_amdgcn_tanhf(x); }
#else
__device__ __forceinline__ float fast_tanh(float x) { return tanhf(x); }
#endif

typedef __attribute__((ext_vector_type(2))) float v2f;
typedef __attribute__((ext_vector_type(8))) float v8f;

// -------------------------------------------------------------------------
// Stage 1: att[n,m] = inv_sqrt_h * sum_k tanh(X[n,k]*Y[m,k])
// 16x16 output tile per 256-thread block; K staged through LDS in 128-chunks.
// -------------------------------------------------------------------------
__global__ __launch_bounds__(256) void att_kernel(const float* __restrict__ X,
                                                  const float* __restrict__ Y,
                                                  float* __restrict__ att)
{
    __shared__ float xs[16][129];   // pad to 129 to avoid bank conflicts
    __shared__ float ys[16][129];
    const int i0 = blockIdx.y * 16;
    const int j0 = blockIdx.x * 16;
    const int tid = threadIdx.x;
    const int tj = tid & 15;
    const int ti = tid >> 4;
    float acc = 0.f;
    for (int k0 = 0; k0 < HDIM; k0 += 128) {
        for (int t = tid; t < 16 * 128; t += 256) {
            const int r = t >> 7, c = t & 127;
            xs[r][c] = X[(i0 + r) * HDIM + k0 + c];
            ys[r][c] = Y[(j0 + r) * HDIM + k0 + c];
        }
        __syncthreads();
        #pragma unroll 8
        for (int c = 0; c < 128; ++c)
            acc += fast_tanh(xs[ti][c] * ys[tj][c]);
        __syncthreads();
    }
    att[(i0 + ti) * N_HYPER + (j0 + tj)] = acc * INV_SQRT_H;
}

// -------------------------------------------------------------------------
// Stage 2a: row softmax (axis = -1) -> A1 (hypo_side_att)
// -------------------------------------------------------------------------
__global__ __launch_bounds__(128) void row_softmax_kernel(const float* __restrict__ att,
                                                          float* __restrict__ A1)
{
    __shared__ float red[128];
    const int row = blockIdx.x;
    const int tid = threadIdx.x;
    const float* a = att + row * N_HYPER;

    float m = -INFINITY;
    for (int c = tid; c < N_HYPER; c += 128) m = fmaxf(m, a[c]);
    red[tid] = m; __syncthreads();
    for (int s = 64; s > 0; s >>= 1) {
        if (tid < s) red[tid] = fmaxf(red[tid], red[tid + s]);
        __syncthreads();
    }
    m = red[0]; __syncthreads();

    float sum = 0.f;
    for (int c = tid; c < N_HYPER; c += 128) sum += __expf(a[c] - m);
    red[tid] = sum; __syncthreads();
    for (int s = 64; s > 0; s >>= 1) {
        if (tid < s) red[tid] += red[tid + s];
        __syncthreads();
    }
    const float inv = 1.f / red[0];
    for (int c = tid; c < N_HYPER; c += 128)
        A1[row * N_HYPER + c] = __expf(a[c] - m) * inv;
}

// -------------------------------------------------------------------------
// Stage 2b: column softmax (axis = 0) -> A2 (hyper_side_att)
// -------------------------------------------------------------------------
__global__ __launch_bounds__(128) void col_softmax_kernel(const float* __restrict__ att,
                                                          float* __restrict__ A2)
{
    __shared__ float red[128];
    const int col = blockIdx.x;
    const int tid = threadIdx.x;

    float m = -INFINITY;
    for (int r = tid; r < N_HYPO; r += 128) m = fmaxf(m, att[r * N_HYPER + col]);
    red[tid] = m; __syncthreads();
    for (int s = 64; s > 0; s >>= 1) {
        if (tid < s) red[tid] = fmaxf(red[tid], red[tid + s]);
        __syncthreads();
    }
    m = red[0]; __syncthreads();

    float sum = 0.f;
    for (int r = tid; r < N_HYPO; r += 128) sum += __expf(att[r * N_HYPER + col] - m);
    red[tid] = sum; __syncthreads();
    for (int s = 64; s > 0; s >>= 1) {
        if (tid < s) red[tid] += red[tid + s];
        __syncthreads();
    }
    const float inv = 1.f / red[0];
    for (int r = tid; r < N_HYPO; r += 128)
        A2[r * N_HYPER + col] = __expf(att[r * N_HYPER + col] - m) * inv;
}

// -------------------------------------------------------------------------
// Stage 3: typicalness vectors.
//   block 0: hyper_typ[m] = colsum_n A1[n,m] / total
//   block 1: hypo_typ[n]  = rowsum_m A2[n,m] / total
// -------------------------------------------------------------------------
__global__ __launch_bounds__(384) void typ_kernel(const float* __restrict__ A1,
                                                  const float* __restrict__ A2,
                                                  float* __restrict__ hyper_typ,
                                                  float* __restrict__ hypo_typ)
{
    __shared__ float red[384];
    __shared__ float inv_total;
    const int tid = threadIdx.x;
    float s = 0.f;
    if (blockIdx.x == 0) {
        for (int n = 0; n < N_HYPO; ++n) s += A1[n * N_HYPER + tid];
    } else {
        for (int m = 0; m < N_HYPER; ++m) s += A2[tid * N_HYPER + m];
    }
    red[tid] = s;
    __syncthreads();
    if (tid == 0) {
        float t = 0.f;
        for (int i = 0; i < 384; ++i) t += red[i];
        inv_total = 1.f / t;
    }
    __syncthreads();
    if (blockIdx.x == 0) hyper_typ[tid] = s * inv_total;
    else                 hypo_typ[tid]  = s * inv_total;
}

// -------------------------------------------------------------------------
// Stage 4: typ-weighted attention collapse.
//   block 0: w_hyper[m] = sum_n hypo_typ[n] * A1[n,m]
//   block 1: w_hypo[n]  = sum_m hyper_typ[m] * A2[n,m]
// -------------------------------------------------------------------------
__global__ __launch_bounds__(384) void wvec_kernel(const float* __restrict__ A1,
                                                   const float* __restrict__ A2,
                                                   const float* __restrict__ hyper_typ,
                                                   const float* __restrict__ hypo_typ,
                                                   float* __restrict__ w_hyper,
                                                   float* __restrict__ w_hypo)
{
    const int tid = threadIdx.x;
    float s = 0.f;
    if (blockIdx.x == 0) {
        for (int n = 0; n < N_HYPO; ++n) s += hypo_typ[n] * A1[n * N_HYPER + tid];
        w_hyper[tid] = s;
    } else {
        for (int m = 0; m < N_HYPER; ++m) s += hyper_typ[m] * A2[tid * N_HYPER + m];
        w_hypo[tid] = s;
    }
}

// -------------------------------------------------------------------------
// Stage 5: prototype GEMVs via V_WMMA_F32_16X16X4_F32 (fp32 exact).
//   proto[d] = sum_k w[k] * E[k,d],  K = 384, D = 768.
// One wave per 16-wide d-tile; 48 tiles x 2 prototypes = 96 waves.
// A tile: w packed into row M=0 (lanes 0 & 16 per the 16x4 f32 layout,
// masked by multiply to keep control flow uniform / EXEC all-ones).
// B tile: 4x16 slice of E (lane = N, vgpr pair = K / K+1, hi half = K+2/K+3).
// D: row M=0 lives in acc[0] on lanes 0..15.
// -------------------------------------------------------------------------
__global__ __launch_bounds__(32) void proto_wmma_kernel(const float* __restrict__ Ehyper,
                                                        const float* __restrict__ Ehypo,
                                                        const float* __restrict__ w_hyper,
                                                        const float* __restrict__ w_hypo,
                                                        float* __restrict__ hyper_proto,
                                                        float* __restrict__ hypo_proto)
{
    const int b = blockIdx.x;       // 0..95
    const int which = b / 48;       // 0 -> hyper prototype, 1 -> hypo prototype
    const int dt = b % 48;
    const float* __restrict__ E = which ? Ehypo : Ehyper;
    const float* __restrict__ w = which ? w_hypo : w_hyper;
    float* __restrict__ p = which ? hypo_proto : hyper_proto;

    const int lane = threadIdx.x;   // 0..31
    const int n  = lane & 15;       // M for A / N for B,D
    const int hi = lane >> 4;       // K half select
    const int d0 = dt * 16;
    const float msk = (n == 0) ? 1.0f : 0.0f;   // only row M=0 of A carries w

    v8f acc = {0.f, 0.f, 0.f, 0.f, 0.f, 0.f, 0.f, 0.f};
    for (int k0 = 0; k0 < N_HYPER; k0 += 4) {
        const int ka = k0 + 2 * hi;
        v2f a;
        a.x = msk * w[ka];
        a.y = msk * w[ka + 1];
        v2f bb;
        bb.x = E[ka * HDIM + d0 + n];
        bb.y = E[(ka + 1) * HDIM + d0 + n];
        acc = __builtin_amdgcn_wmma_f32_16x16x4_f32(
            /*neg_a=*/false, a, /*neg_b=*/false, bb,
            /*c_mod=*/(short)0, acc, /*reuse_a=*/false, /*reuse_b=*/false);
    }
    if (lane < 16) p[d0 + lane] = acc[0];
}

// -------------------------------------------------------------------------
// Stage 6: classifier head. One block per label.
// feats = [hp, qp, hp-qp, hp*qp]  (hp = hyper_proto, qp = hypo_proto)
// -------------------------------------------------------------------------
__global__ __launch_bounds__(256) void cls_kernel(const float* __restrict__ W,
                                                  const float* __restrict__ bias,
                                                  const float* __restrict__ hp,
                                                  const float* __restrict__ qp,
                                                  float* __restrict__ out)
{
    __shared__ float red[256];
    const int l = blockIdx.x;
    const int tid = threadIdx.x;
    float s = 0.f;
    for (int f = tid; f < 4 * HDIM; f += 256) {
        float feat;
        if (f < HDIM)            feat = hp[f];
        else if (f < 2 * HDIM)   feat = qp[f - HDIM];
        else if (f < 3 * HDIM)   feat = hp[f - 2 * HDIM] - qp[f - 2 * HDIM];
        else                     feat = hp[f - 3 * HDIM] * qp[f - 3 * HDIM];
        s += W[l * 4 * HDIM + f] * feat;
    }
    red[tid] = s; __syncthreads();
    for (int st = 128; st > 0; st >>= 1) {
        if (tid < st) red[tid] += red[tid + st];
        __syncthreads();
    }
    if (tid == 0) out[l] = red[0] + bias[l];
}

// -------------------------------------------------------------------------
extern "C" void kernel_launch(void* const* d_in, const int* in_sizes, int n_in,
                              void* d_out, int out_size, void* d_ws, size_t ws_size,
                              hipStream_t stream)
{
    const float* hypo  = (const float*)d_in[0];   // [384, 768]
    const float* hyper = (const float*)d_in[1];   // [384, 768]
    const float* Wcls  = (const float*)d_in[2];   // [3, 3072]
    const float* bcls  = (const float*)d_in[3];   // [3]
    float* out = (float*)d_out;                   // [3]

    float* ws = (float*)d_ws;
    float* att          = ws;                               // 384*384
    float* A1           = att + N_HYPO * N_HYPER;           // 384*384
    float* A2           = A1  + N_HYPO * N_HYPER;           // 384*384
    float* hyper_typ    = A2  + N_HYPO * N_HYPER;           // 384
    float* hypo_typ     = hyper_typ + N_HYPER;              // 384
    float* w_hyper      = hypo_typ  + N_HYPO;               // 384
    float* w_hypo       = w_hyper   + N_HYPER;              // 384
    float* hyper_proto  = w_hypo    + N_HYPO;               // 768
    float* hypo_proto   = hyper_proto + HDIM;               // 768

    att_kernel<<<dim3(N_HYPER / 16, N_HYPO / 16), 256, 0, stream>>>(hypo, hyper, att);
    row_softmax_kernel<<<N_HYPO, 128, 0, stream>>>(att, A1);
    col_softmax_kernel<<<N_HYPER, 128, 0, stream>>>(att, A2);
    typ_kernel<<<2, 384, 0, stream>>>(A1, A2, hyper_typ, hypo_typ);
    wvec_kernel<<<2, 384, 0, stream>>>(A1, A2, hyper_typ, hypo_typ, w_hyper, w_hypo);
    proto_wmma_kernel<<<96, 32, 0, stream>>>(hyper, hypo, w_hyper, w_hypo,
                                             hyper_proto, hypo_proto);
    cls_kernel<<<NLAB, 256, 0, stream>>>(Wcls, bcls, hyper_proto, hypo_proto, out);
}